// mv_embedding_10402410791516
// MI455X (gfx1250) — compile-verified
//
#include <hip/hip_runtime.h>

// ---------------------------------------------------------------------------
// Fused MLP pipeline for MI455X (gfx1250, wave32, WMMA + async-to-LDS).
//   B=512, K=128, IN=256, E=256, H=1280, OUT=128
// All three matmuls are C[M,N] = A[M,Kd] * W[N,Kd]^T with K-contiguous
// operands -> one templated WMMA GEMM serves all of them.
// Roofline: ~240MB traffic vs ~52 GFLOP -> HBM-bound (~5 TFLOP/s needed),
// so full f32 precision via V_WMMA_F32_16X16X4_F32 is the right choice.
// Staging uses GLOBAL_LOAD_ASYNC_TO_LDS_B128 (ASYNCcnt) with LDS double
// buffering so the 168MB W1 stream overlaps the WMMA pipeline.
// ---------------------------------------------------------------------------

typedef __attribute__((ext_vector_type(2))) float v2f;
typedef __attribute__((ext_vector_type(4))) float v4f;
typedef __attribute__((ext_vector_type(8))) float v8f;

#define BM 128
#define BN 128
#define BK 16
#define LDSW 20            // floats per LDS row: BK + 4 pad
                           // 80B rows: 16B-aligned b128 slots; 20*r mod 64
                           // hits all banks -> conflict-free ds_load_b64 frags

#define DIM_B   512
#define DIM_K   128
#define DIM_IN  256
#define DIM_E   256
#define DIM_H   1280
#define DIM_OUT 128
#define DIM_EK  (DIM_E * DIM_K)   // 32768

enum EpiMode { EPI_EMBED_T = 0, EPI_SIGMOID = 1, EPI_PLAIN = 2 };

// Async 16B global->LDS copy (per enabled lane). dst: wave-relative LDS byte
// address (generic LDS pointer low 32 bits), src: 64-bit global address.
__device__ __forceinline__ void async_copy_b128(unsigned lds_byte, const float* g) {
  asm volatile("global_load_async_to_lds_b128 %0, %1, off"
               :: "v"(lds_byte), "v"((unsigned long long)(uintptr_t)g)
               : "memory");
}
__device__ __forceinline__ void wait_async0() {
  asm volatile("s_wait_asynccnt 0x0" ::: "memory");
}

// ---------------------------------------------------------------------------
// Templated WMMA GEMM:  C = epilogue(A * W^T + bias)
//   A : [M, Kd] row-major (optionally BN-normalized per column-block of 128)
//   W : [N, Kd] row-major
//   256 threads = 8 waves, tile BM x BN, waves in 4x2 grid, each wave owns a
//   32x64 patch = 2x4 fragments of 16x16 (v8f accumulators).
// ---------------------------------------------------------------------------
template <int EPI, bool NORM_A>
__global__ __launch_bounds__(256)
void gemm_wmma_kernel(const float* __restrict__ A,
                      const float* __restrict__ W,
                      const float* __restrict__ bias,
                      float* __restrict__ C,
                      const float* __restrict__ scale,
                      const float* __restrict__ shift,
                      int M, int N, int Kd) {
  __shared__ float As[2][BM * LDSW];   // double-buffered: 2 x 10KB
  __shared__ float Bs[2][BN * LDSW];   // double-buffered: 2 x 10KB  (40KB total)

  const int tid  = threadIdx.x;       // 0..255
  const int wave = tid >> 5;          // 0..7
  const int lane = tid & 31;
  const int half = lane >> 4;         // lane group: K split per ISA frag layout
  const int l16  = lane & 15;

  const int wm = wave & 3;            // 0..3 -> M offset wm*32
  const int wn = wave >> 2;           // 0..1 -> N offset wn*64

  const int tileN = blockIdx.x * BN;
  const int tileM = blockIdx.y * BM;

  v8f acc[2][4];
#pragma unroll
  for (int i = 0; i < 2; ++i)
#pragma unroll
    for (int j = 0; j < 4; ++j)
      acc[i][j] = (v8f)(0.0f);

  // Staging map: 256 threads x float4, 4 threads per 16-float row,
  // 64 rows per pass, 2 passes per 128-row tile.
  const int srow = tid >> 2;          // 0..63
  const int scol = (tid & 3) * 4;     // 0,4,8,12

  const unsigned asBase = (unsigned)(uintptr_t)&As[0][0];
  const unsigned bsBase = (unsigned)(uintptr_t)&Bs[0][0];
  const unsigned ldsSlot = (unsigned)((srow * LDSW + scol) * sizeof(float));
  const unsigned bufBytes = (unsigned)(BM * LDSW * sizeof(float));

  const int nK = Kd / BK;

  // Issue a K-tile's async loads into buffer `buf`.
  auto issue_tile = [&](int kt, int buf) {
    const int kk0 = kt * BK;
#pragma unroll
    for (int p = 0; p < 2; ++p) {
      const int r = srow + p * 64;
      const unsigned loff = ldsSlot + (unsigned)(p * 64 * LDSW * sizeof(float));
      async_copy_b128(asBase + buf * bufBytes + loff,
                      A + (size_t)(tileM + r) * Kd + kk0 + scol);
      async_copy_b128(bsBase + buf * bufBytes + loff,
                      W + (size_t)(tileN + r) * Kd + kk0 + scol);
    }
  };

  issue_tile(0, 0);   // prologue: tile 0 in flight

  for (int kt = 0; kt < nK; ++kt) {
    wait_async0();        // this wave's tile-kt copies have landed in LDS
    __syncthreads();      // all waves' tile-kt data visible; all waves are
                          // also done reading buf[(kt-1)&1] -> safe to refill
    if (kt + 1 < nK) issue_tile(kt + 1, (kt + 1) & 1);

    const float* as = &As[kt & 1][0];
    const float* bs = &Bs[kt & 1][0];

    // BatchNorm fold for FC1: kk0 is 16-aligned and 16 | 128, so channel
    // e = kk/128 is constant across the K-tile -> scalar scale/shift.
    float nsc = 1.0f, nsh = 0.0f;
    if (NORM_A) {
      const int e = (kt * BK) >> 7;
      nsc = scale[e];
      nsh = shift[e];
    }

    // 4 k-steps of 4; 8 WMMAs per step per wave.
    // ISA 32-bit A 16x4 layout: VGPR0 = K {0 | 2}, VGPR1 = K {1 | 3} across
    // lane halves -> per-lane float2 at k = ks*4 + 2*half.
#pragma unroll
    for (int ks = 0; ks < BK / 4; ++ks) {
      const int kf = ks * 4 + half * 2;
      v2f afrag[2], bfrag[4];
#pragma unroll
      for (int i = 0; i < 2; ++i) {
        const int m = wm * 32 + i * 16 + l16;
        afrag[i] = *(const v2f*)(as + m * LDSW + kf);   // ds_load_b64
        if (NORM_A) afrag[i] = afrag[i] * nsc + nsh;    // fused BatchNorm
      }
#pragma unroll
      for (int j = 0; j < 4; ++j) {
        const int n = wn * 64 + j * 16 + l16;
        bfrag[j] = *(const v2f*)(bs + n * LDSW + kf);
      }
#pragma unroll
      for (int i = 0; i < 2; ++i)
#pragma unroll
        for (int j = 0; j < 4; ++j)
          acc[i][j] = __builtin_amdgcn_wmma_f32_16x16x4_f32(
              false, afrag[i], false, bfrag[j],
              (short)0, acc[i][j], false, false);
    }
    __syncthreads();   // all waves done with buf[kt&1] before it is refilled
  }

  // Epilogue. C/D fragment layout: VGPR r -> M = r + 8*half, N = lane%16.
#pragma unroll
  for (int i = 0; i < 2; ++i) {
    const int gm0 = tileM + wm * 32 + i * 16 + half * 8;
#pragma unroll
    for (int j = 0; j < 4; ++j) {
      const int gn = tileN + wn * 64 + j * 16 + l16;
      const float bv = bias[gn];
#pragma unroll
      for (int r = 0; r < 8; ++r) {
        const int m = gm0 + r;
        float v = acc[i][j][r] + bv;
        if (EPI == EPI_SIGMOID) v = 1.0f / (1.0f + __expf(-v));
        if (EPI == EPI_EMBED_T) {
          // m encodes (b,k) of flattened [B,K]; n is channel e. Store
          // transposed as [b][e*128 + k] so FC1's A operand is row-major.
          const int b = m >> 7, k = m & (DIM_K - 1);
          C[(size_t)b * DIM_EK + (size_t)gn * DIM_K + k] = v;
        } else {
          C[(size_t)m * N + gn] = v;
        }
      }
    }
  }
}

// ---------------------------------------------------------------------------
// BatchNorm statistics over (B,K) per channel e, folded with gamma/beta into
// scale/shift. yt layout: [b][e*128 + k]. One 256-thread block per channel.
// ---------------------------------------------------------------------------
__global__ __launch_bounds__(256)
void bn_stats_kernel(const float* __restrict__ yt,
                     const float* __restrict__ gamma,
                     const float* __restrict__ beta,
                     float* __restrict__ scale,
                     float* __restrict__ shift) {
  const int e   = blockIdx.x;          // 0..255
  const int tid = threadIdx.x;         // 0..255
  const int NSAMP = DIM_B * DIM_K;     // 65536

  float s = 0.0f, s2 = 0.0f;
  for (int i = tid; i < NSAMP; i += 256) {
    const int b = i >> 7, k = i & (DIM_K - 1);
    const float v = yt[(size_t)b * DIM_EK + (size_t)e * DIM_K + k];
    s += v;
    s2 += v * v;
  }

  __shared__ float rs[256], rs2[256];
  rs[tid] = s;
  rs2[tid] = s2;
  __syncthreads();
  for (int off = 128; off > 0; off >>= 1) {
    if (tid < off) {
      rs[tid] += rs[tid + off];
      rs2[tid] += rs2[tid + off];
    }
    __syncthreads();
  }

  if (tid == 0) {
    const float inv_n = 1.0f / (float)NSAMP;
    const float mean = rs[0] * inv_n;
    const float var  = rs2[0] * inv_n - mean * mean;   // population var (jnp.var)
    const float rstd = rsqrtf(var + 1e-5f);
    const float sc = gamma[e] * rstd;
    scale[e] = sc;
    shift[e] = beta[e] - mean * sc;
  }
}

// ---------------------------------------------------------------------------
// Launch: 4 kernels on `stream`.
//   ws layout: yt [B][E*K] (64MB) | z [B][H] (2.5MB) | scale[E] | shift[E]
// ---------------------------------------------------------------------------
extern "C" void kernel_launch(void* const* d_in, const int* in_sizes, int n_in,
                              void* d_out, int out_size, void* d_ws, size_t ws_size,
                              hipStream_t stream) {
  (void)in_sizes; (void)n_in; (void)out_size; (void)ws_size;

  const float* x     = (const float*)d_in[0];   // [B,K,IN]
  const float* W_emb = (const float*)d_in[1];   // [E,IN]
  const float* b_emb = (const float*)d_in[2];   // [E]
  const float* gamma = (const float*)d_in[3];   // [E]
  const float* beta  = (const float*)d_in[4];   // [E]
  const float* W1    = (const float*)d_in[5];   // [H, E*K]
  const float* b1    = (const float*)d_in[6];   // [H]
  const float* W2    = (const float*)d_in[7];   // [OUT, H]
  const float* b2    = (const float*)d_in[8];   // [OUT]
  float* out = (float*)d_out;                   // [B, OUT]

  char* ws = (char*)d_ws;
  float* yt    = (float*)ws;                                              // B*E*K
  float* z     = (float*)(ws + (size_t)DIM_B * DIM_EK * sizeof(float));   // B*H
  float* scale = (float*)((char*)z + (size_t)DIM_B * DIM_H * sizeof(float));
  float* shift = scale + DIM_E;

  // 1) Embedding GEMM: [65536,256] x [256,256]^T -> yt stored as [b][e*128+k]
  gemm_wmma_kernel<EPI_EMBED_T, false>
      <<<dim3(DIM_E / BN, (DIM_B * DIM_K) / BM), 256, 0, stream>>>(
          x, W_emb, b_emb, yt, nullptr, nullptr,
          DIM_B * DIM_K, DIM_E, DIM_IN);

  // 2) BatchNorm stats -> folded scale/shift per channel
  bn_stats_kernel<<<DIM_E, 256, 0, stream>>>(yt, gamma, beta, scale, shift);

  // 3) FC1 (dominant, W1 = 168MB streamed): fused BN-on-load + sigmoid
  gemm_wmma_kernel<EPI_SIGMOID, true>
      <<<dim3(DIM_H / BN, DIM_B / BM), 256, 0, stream>>>(
          yt, W1, b1, z, scale, shift,
          DIM_B, DIM_H, DIM_EK);

  // 4) FC2: [512,1280] x [128,1280]^T -> d_out
  gemm_wmma_kernel<EPI_PLAIN, false>
      <<<dim3(DIM_OUT / BN, DIM_B / BM), 256, 0, stream>>>(
          z, W2, b2, out, nullptr, nullptr,
          DIM_B, DIM_OUT, DIM_H);
}